// SPP_9182640079004
// MI455X (gfx1250) — compile-verified
//
#include <hip/hip_runtime.h>

// SPP: out = concat(x, maxpool5(x), maxpool9(x), maxpool13(x)) on channels.
// x: (32, 512, 32, 32) fp32 -> out: (32, 2048, 32, 32) fp32.
// HBM-bound: 64 MiB in + 256 MiB out ~= 320 MiB @ 23.3 TB/s => ~14 us floor.
// One workgroup per (b,c) plane. Plane in LDS with -inf halos => branch-free
// separable passes. Cascade: mp9 = mp5(mp5), mp13 = mp5(mp5(mp5)) (exact for
// stride-1 / -inf padding). Input via gfx1250 async global->LDS B128; outputs
// via non-temporal B128 stores (256 MiB output > 192 MB L2, never re-read).

typedef float f4 __attribute__((ext_vector_type(4)));
typedef int   v4i __attribute__((__vector_size__(4 * sizeof(int))));

#define NEG_INF (-__builtin_huge_valf())
#define AS1 __attribute__((address_space(1)))
#define AS3 __attribute__((address_space(3)))

// Padded tile: 36 rows (2 halo top/bottom) x 40 cols (4 left pad for 16B
// alignment, 4 right pad; only 2 of each needed as halo). 1440 floats.
#define PW   40
#define PBUF 1440
#define PIDX(i, j) ((((i) + 2) * PW) + 4 + (j))

#if __has_builtin(__builtin_amdgcn_global_load_async_to_lds_b128) && \
    __has_builtin(__builtin_amdgcn_s_wait_asynccnt)
#define SPP_ASYNC 1
#else
#define SPP_ASYNC 0
#endif

// Horizontal 5-max over one row chunk (4 consecutive pixels). Branch-free:
// halo columns hold -inf. Writes interior of dst only (dst row-halo stays -inf).
__device__ __forceinline__ void hmax5(const float* __restrict__ src,
                                      float* __restrict__ dst, int i, int j0) {
    const float* r = src + PIDX(i, 0);
    float w[8];
#pragma unroll
    for (int k = 0; k < 8; ++k) w[k] = r[j0 - 2 + k];
    f4 o;
    o.x = fmaxf(fmaxf(w[0], w[1]), fmaxf(w[2], fmaxf(w[3], w[4])));
    o.y = fmaxf(fmaxf(w[1], w[2]), fmaxf(w[3], fmaxf(w[4], w[5])));
    o.z = fmaxf(fmaxf(w[2], w[3]), fmaxf(w[4], fmaxf(w[5], w[6])));
    o.w = fmaxf(fmaxf(w[3], w[4]), fmaxf(w[5], fmaxf(w[6], w[7])));
    *(f4*)(dst + PIDX(i, j0)) = o;
}

// Vertical 5-max, float4 per thread, unconditional (row halo holds -inf).
__device__ __forceinline__ f4 vmax5(const float* __restrict__ src, int i, int j0) {
    f4 acc = *(const f4*)(src + PIDX(i - 2, j0));
#pragma unroll
    for (int dr = -1; dr <= 2; ++dr) {
        const f4 v = *(const f4*)(src + PIDX(i + dr, j0));
        acc.x = fmaxf(acc.x, v.x);
        acc.y = fmaxf(acc.y, v.y);
        acc.z = fmaxf(acc.z, v.z);
        acc.w = fmaxf(acc.w, v.w);
    }
    return acc;
}

__global__ __launch_bounds__(256) void spp_kernel(const float* __restrict__ x,
                                                  float* __restrict__ out) {
    __shared__ __align__(16) float sA[PBUF];
    __shared__ __align__(16) float sB[PBUF];
    __shared__ __align__(16) float sC[PBUF];

    const int tid   = threadIdx.x;
    const int plane = blockIdx.x;          // b*512 + c
    const int b     = plane >> 9;
    const int c     = plane & 511;
    const int q0    = tid << 2;            // 4 consecutive pixels per thread
    const int i     = q0 >> 5;
    const int j0    = q0 & 31;

    // ---- fill all buffers (incl. halos) with -inf; halos stay -inf forever
    {
        const f4 n4 = {NEG_INF, NEG_INF, NEG_INF, NEG_INF};
        for (int t = tid; t < PBUF / 4; t += 256) {
            ((f4*)sA)[t] = n4;
            ((f4*)sB)[t] = n4;
            ((f4*)sC)[t] = n4;
        }
    }
    __syncthreads();   // init visible before async DMA overwrites sA interior

    const float* src = x + ((size_t)plane << 10);

    // ---- load plane interior into LDS (gfx1250 async global->LDS) ----
#if SPP_ASYNC
    __builtin_amdgcn_global_load_async_to_lds_b128(
        (AS1 v4i*)(src + q0),
        (AS3 v4i*)(sA + PIDX(i, j0)),
        0, 0);
    __builtin_amdgcn_s_wait_asynccnt(0);
#else
    *(f4*)(sA + PIDX(i, j0)) = *(const f4*)(src + q0);
#endif
    __syncthreads();

    float* out0 = out + (((size_t)b * 2048 + 0 * 512 + c) << 10);
    float* out1 = out + (((size_t)b * 2048 + 1 * 512 + c) << 10);
    float* out2 = out + (((size_t)b * 2048 + 2 * 512 + c) << 10);
    float* out3 = out + (((size_t)b * 2048 + 3 * 512 + c) << 10);

    // section 0: passthrough x
    {
        f4 v = *(const f4*)(sA + PIDX(i, j0));
        __builtin_nontemporal_store(v, (f4*)(out0 + q0));
    }

    // p5 = v5(h5(x))
    hmax5(sA, sB, i, j0);
    __syncthreads();
    {
        f4 v = vmax5(sB, i, j0);
        *(f4*)(sC + PIDX(i, j0)) = v;
        __builtin_nontemporal_store(v, (f4*)(out1 + q0));
    }
    __syncthreads();

    // p9 = mp5(p5)
    hmax5(sC, sB, i, j0);
    __syncthreads();
    {
        f4 v = vmax5(sB, i, j0);
        *(f4*)(sA + PIDX(i, j0)) = v;
        __builtin_nontemporal_store(v, (f4*)(out2 + q0));
    }
    __syncthreads();

    // p13 = mp5(p9)
    hmax5(sA, sB, i, j0);
    __syncthreads();
    {
        f4 v = vmax5(sB, i, j0);
        __builtin_nontemporal_store(v, (f4*)(out3 + q0));
    }
}

extern "C" void kernel_launch(void* const* d_in, const int* in_sizes, int n_in,
                              void* d_out, int out_size, void* d_ws, size_t ws_size,
                              hipStream_t stream) {
    (void)in_sizes; (void)n_in; (void)d_ws; (void)ws_size; (void)out_size;
    const float* x = (const float*)d_in[0];
    float* out = (float*)d_out;
    // 32 batches * 512 channels = 16384 planes, one block each.
    dim3 grid(32 * 512);
    dim3 block(256);
    spp_kernel<<<grid, block, 0, stream>>>(x, out);
}